// MultiScale_GraphConv_43336220016694
// MI455X (gfx1250) — compile-verified
//
#include <hip/hip_runtime.h>
#include <hip/hip_bf16.h>
#include <stddef.h>

// ---- problem constants (from reference setup_inputs) ----
#define NN   32
#define CC   64
#define TT   128
#define VV   25
#define NSS  13
#define CO   384
#define KDIM (NSS*CC)   // 832
#define COLS (TT*VV)    // 3200
#define NB   4          // n-chunk for support buffer

typedef __attribute__((ext_vector_type(2))) float v2f;
typedef __attribute__((ext_vector_type(8))) float v8f;
typedef __attribute__((ext_vector_type(4))) unsigned int tdm_u4;
typedef __attribute__((ext_vector_type(8))) int tdm_i8;
typedef __attribute__((ext_vector_type(4))) int tdm_i4;

// =====================================================================
// TDM helper: 2D tile load global->LDS via Tensor Data Mover.
//   lds_addr   : raw LDS byte offset of destination
//   gptr       : global byte address of tile start
//   tile0/tile1: tile width/height in elements (x contiguous)
//   stride0    : row stride in elements
//   d0flags    : group1 dword0 (data_size | pad config), workgroup_mask=0
// =====================================================================
__device__ __forceinline__ void tdm_load_2d(unsigned lds_addr, const void* gptr,
                                            unsigned tile0, unsigned tile1,
                                            unsigned long long stride0,
                                            unsigned d0flags) {
    unsigned long long ga = (unsigned long long)(size_t)gptr;
    tdm_u4 g0;
    g0.x = 1u;                                           // count=1, user mode
    g0.y = lds_addr;                                     // lds_addr[63:32]
    g0.z = (unsigned)(ga & 0xFFFFFFFFu);                 // global_addr[95:64]
    g0.w = (unsigned)((ga >> 32) & 0x1FFFFFFu) | (2u << 30); // addr[56:32]|type=2
    const unsigned dim0 = tile0, dim1 = tile1;           // exact-fit: no OOB
    tdm_i8 g1;
    g1[0] = (int)d0flags;                                // mask=0|data_size|pad
    g1[1] = (int)((dim0 & 0xFFFFu) << 16);               // abar=0 | dim0.lo16
    g1[2] = (int)(((dim0 >> 16) & 0xFFFFu) | ((dim1 & 0xFFFFu) << 16));
    g1[3] = (int)(((dim1 >> 16) & 0xFFFFu) | ((tile0 & 0xFFFFu) << 16));
    g1[4] = (int)(tile1 & 0xFFFFu);                      // tile1 | tile2=0
    g1[5] = (int)(unsigned)(stride0 & 0xFFFFFFFFull);    // stride0[31:0]
    g1[6] = (int)(unsigned)((stride0 >> 32) & 0xFFFFull);// stride0.hi | stride1=0
    g1[7] = 0;
    tdm_i4 gz = {0, 0, 0, 0};                            // 2D: groups 2/3 unused
#if defined(__clang_major__) && (__clang_major__ >= 23)
    tdm_i8 gz8 = {0, 0, 0, 0, 0, 0, 0, 0};
    __builtin_amdgcn_tensor_load_to_lds(g0, g1, gz, gz, gz8, 0);
#else
    __builtin_amdgcn_tensor_load_to_lds(g0, g1, gz, gz, 0);
#endif
}

// d0flags builders: data_size=2 (4 bytes) at bits 17:16; pad_enable bit 20;
// pad_interval bits 24:22 (code k -> 2^(k+1) dwords); pad_amount bits 31:25
// (code a -> a+1 dwords).
#define TDM_F32           (2u << 16)
#define TDM_PAD(ic, ac)   ((1u << 20) | ((unsigned)(ic) << 22) | ((unsigned)(ac) << 25))

// =====================================================================
// K0: per-(n,t) dynamic adjacency  A_norm = D^-1/2 exp(-d2) D^-1/2,
//     accumulated over t into Bsum[n] (25x25) with atomics.
// =====================================================================
__global__ __launch_bounds__(256) void k0_adj(const float* __restrict__ xx,
                                              float* __restrict__ Bsum) {
    const int n = blockIdx.x, t = blockIdx.y;
    __shared__ float feat[VV][CC + 1];
    __shared__ float gram[VV][VV];
    __shared__ float Amat[VV][VV];
    __shared__ float dinv[VV];
    const int tid = threadIdx.x;

    for (int i = tid; i < VV * CC; i += 256) {
        int c = i / VV, v = i % VV;
        feat[v][c] = xx[(((size_t)n * CC + c) * TT + t) * VV + v];
    }
    __syncthreads();

    for (int p = tid; p < VV * (VV + 1) / 2; p += 256) {
        int v = 0, pp = p;
        while (pp >= VV - v) { pp -= (VV - v); v++; }
        int u = v + pp;
        float s = 0.f;
        #pragma unroll 8
        for (int c = 0; c < CC; c++) s += feat[v][c] * feat[u][c];
        gram[v][u] = s; gram[u][v] = s;
    }
    __syncthreads();

    for (int p = tid; p < VV * VV; p += 256) {
        int v = p / VV, u = p % VV;
        float d2 = fmaxf(gram[v][v] + gram[u][u] - 2.f * gram[v][u], 0.f);
        Amat[v][u] = expf(-d2);
    }
    __syncthreads();

    if (tid < VV) {
        float s = 0.f;
        for (int u = 0; u < VV; u++) s += Amat[tid][u];
        dinv[tid] = rsqrtf(s);
    }
    __syncthreads();

    for (int p = tid; p < VV * VV; p += 256) {
        int v = p / VV, u = p % VV;
        atomicAdd(&Bsum[n * VV * VV + p], dinv[v] * dinv[u] * Amat[v][u]);
    }
}

// =====================================================================
// K1: AC[n, s*25+v, u] = Bsum[n,v,u] * (s==0 ? 1 : A_powers[(s-1)*25+v, u])
// =====================================================================
__global__ void k1_ac(const float* __restrict__ Bsum,
                      const float* __restrict__ Apow,
                      float* __restrict__ AC) {
    int idx = blockIdx.x * 256 + threadIdx.x;
    const int total = NN * NSS * VV * VV;
    if (idx >= total) return;
    int u = idx % VV;
    int v = (idx / VV) % VV;
    int s = (idx / (VV * VV)) % NSS;
    int n = idx / (VV * VV * NSS);
    float b = Bsum[(n * VV + v) * VV + u];
    float m = (s == 0) ? 1.0f : Apow[((s - 1) * VV + v) * VV + u];
    AC[((size_t)n * NSS + s) * (VV * VV) + v * VV + u] = b * m;
}

// =====================================================================
// K2: support GEMM via f32 WMMA.
//   block = (c, s, n_local); S[(s*64+c),(f*25+v)] = sum_u x[c,f,u]*AC[s*25+v,u]
// =====================================================================
__global__ __launch_bounds__(256) void k2_support(const float* __restrict__ x,
                                                  const float* __restrict__ AC,
                                                  float* __restrict__ S, int n0) {
    const int c  = blockIdx.x;
    const int s  = blockIdx.y;
    const int nl = blockIdx.z;
    const int n  = n0 + nl;
    __shared__ float ldsX[TT][VV + 1];
    __shared__ float ldsB[28][33];
    const int tid = threadIdx.x;

    const float* xp = x + ((size_t)n * CC + c) * TT * VV;
    for (int i = tid; i < TT * VV; i += 256) ldsX[i / VV][i % VV] = xp[i];
    const float* acp = AC + ((size_t)n * NSS + s) * (VV * VV);
    for (int i = tid; i < 28 * 32; i += 256) {
        int u = i / 32, v = i % 32;
        ldsB[u][v] = (u < VV && v < VV) ? acp[v * VV + u] : 0.f;
    }
    __syncthreads();

    const int wid = tid >> 5, lane = tid & 31;
    const int lm = lane & 15;
    const bool hi = lane >= 16;
    const int fbase = wid * 16;

    v8f acc0 = {0.f,0.f,0.f,0.f,0.f,0.f,0.f,0.f};
    v8f acc1 = acc0;

    #pragma unroll
    for (int kb = 0; kb < 28; kb += 4) {
        const int u0 = kb + (hi ? 2 : 0);
        const int f  = fbase + lm;
        v2f a, b0, b1;
        a.x  = (u0     < VV) ? ldsX[f][u0]     : 0.f;
        a.y  = (u0 + 1 < VV) ? ldsX[f][u0 + 1] : 0.f;
        b0.x = ldsB[u0][lm];      b0.y = ldsB[u0 + 1][lm];
        b1.x = ldsB[u0][16 + lm]; b1.y = ldsB[u0 + 1][16 + lm];
        acc0 = __builtin_amdgcn_wmma_f32_16x16x4_f32(false, a, false, b0,
                                                     (short)0, acc0, false, false);
        acc1 = __builtin_amdgcn_wmma_f32_16x16x4_f32(false, a, false, b1,
                                                     (short)0, acc1, false, false);
    }

    float* sp = S + ((size_t)nl * KDIM + (s * CC + c)) * COLS;
    #pragma unroll
    for (int i = 0; i < 8; i++) {
        int f = fbase + (hi ? i + 8 : i);
        sp[f * VV + lm] = acc0[i];
        if (16 + lm < VV) sp[f * VV + 16 + lm] = acc1[i];
    }
}

// =====================================================================
// K3: main MLP GEMM  Y[n] = W(384x832) x S[nl](832x3200), f32 WMMA.
//   128x128 block tile, BK=16, double-buffered LDS fed by TDM
//   (tensor_load_to_lds), one barrier per K-iteration.
//   LDS layouts (TDM row padding):
//     lA[buf]: 128 rows x (16+4) floats, [m][k]   (pad 4 dw / 16 dw row)
//     lB[buf]: 16 rows x (128+8) floats, [k][n]   (pad 8 dw / 128 dw row)
// =====================================================================
#define BM 128
#define BN 128
#define BK 16
#define LAS (BK + 4)     // 20 floats per A row
#define LBS (BN + 8)     // 136 floats per B row
#define NIT (KDIM / BK)  // 52

__global__ __launch_bounds__(256) void k3_gemm(const float* __restrict__ W,
                                               const float* __restrict__ S,
                                               float* __restrict__ Y, int n0) {
    const int bn = blockIdx.x;          // 0..24
    const int bm = blockIdx.y;          // 0..2
    const int nl = blockIdx.z;
    const int n  = n0 + nl;
    __shared__ float lA[2][BM * LAS];   // 2 x 10240 B
    __shared__ float lB[2][BK * LBS];   // 2 x  8704 B

    const int tid = threadIdx.x, lane = tid & 31;
    const int wid = tid >> 5;
    const int wm = wid >> 2, wn = wid & 3;   // 2 x 4 wave grid
    const int lm = lane & 15;
    const bool hi = lane >= 16;

    // wave-uniform "am I wave 0" -> scalar branch (TDM ignores EXEC, so the
    // issue must be reached by exactly one wave via scalar control flow)
    const bool wf0 = (__builtin_amdgcn_readfirstlane(tid) >> 5) == 0;

    const v8f vz = {0.f,0.f,0.f,0.f,0.f,0.f,0.f,0.f};
    v8f acc[4][2];
    #pragma unroll
    for (int i = 0; i < 4; i++) { acc[i][0] = vz; acc[i][1] = vz; }

    const float* Wp = W + (size_t)(bm * BM) * KDIM;          // tile row origin
    const float* Sp = S + (size_t)nl * KDIM * COLS + (size_t)bn * BN;

    // prologue: fill buffer 0
    if (wf0) {
        tdm_load_2d((unsigned)(size_t)&lA[0][0], Wp,
                    BK, BM, (unsigned long long)KDIM, TDM_F32 | TDM_PAD(3, 3));
        tdm_load_2d((unsigned)(size_t)&lB[0][0], Sp,
                    BN, BK, (unsigned long long)COLS, TDM_F32 | TDM_PAD(6, 7));
        __builtin_amdgcn_s_wait_tensorcnt(0);
    }
    __syncthreads();

    for (int it = 0; it < NIT; ++it) {
        const int cur = it & 1, nxt = cur ^ 1;

        // issue next-tile DMAs; they overlap with this iteration's WMMAs
        if (it + 1 < NIT && wf0) {
            const int k1 = (it + 1) * BK;
            tdm_load_2d((unsigned)(size_t)&lA[nxt][0], Wp + k1,
                        BK, BM, (unsigned long long)KDIM, TDM_F32 | TDM_PAD(3, 3));
            tdm_load_2d((unsigned)(size_t)&lB[nxt][0], Sp + (size_t)k1 * COLS,
                        BN, BK, (unsigned long long)COLS, TDM_F32 | TDM_PAD(6, 7));
        }

        const float* la = &lA[cur][0];
        const float* lb = &lB[cur][0];
        #pragma unroll
        for (int ks = 0; ks < 4; ks++) {
            const int kk = ks * 4 + (hi ? 2 : 0);
            v2f af[4], bf[2];
            #pragma unroll
            for (int tm = 0; tm < 4; tm++) {
                int m = wm * 64 + tm * 16 + lm;
                af[tm].x = la[m * LAS + kk];
                af[tm].y = la[m * LAS + kk + 1];
            }
            #pragma unroll
            for (int tn = 0; tn < 2; tn++) {
                int nn = wn * 32 + tn * 16 + lm;
                bf[tn].x = lb[kk * LBS + nn];
                bf[tn].y = lb[(kk + 1) * LBS + nn];
            }
            #pragma unroll
            for (int tm = 0; tm < 4; tm++)
                #pragma unroll
                for (int tn = 0; tn < 2; tn++)
                    acc[tm][tn] = __builtin_amdgcn_wmma_f32_16x16x4_f32(
                        false, af[tm], false, bf[tn], (short)0, acc[tm][tn],
                        false, false);
        }

        if (wf0) __builtin_amdgcn_s_wait_tensorcnt(0);  // next buffer landed
        __syncthreads();                                 // all waves see it
    }

    // store (b_mlp omitted: per-channel bias cancels exactly under BN mean-sub)
    float* yp = Y + ((size_t)n * CO + bm * BM) * COLS + (size_t)bn * BN;
    #pragma unroll
    for (int tm = 0; tm < 4; tm++)
        #pragma unroll
        for (int tn = 0; tn < 2; tn++) {
            int mrow = wm * 64 + tm * 16;
            int ncol = wn * 32 + tn * 16 + lm;
            #pragma unroll
            for (int i = 0; i < 8; i++) {
                int r = mrow + (hi ? i + 8 : i);
                yp[(size_t)r * COLS + ncol] = acc[tm][tn][i];
            }
        }
}

// =====================================================================
// K4: per-channel BN batch stats -> scale/shift
// =====================================================================
__global__ __launch_bounds__(256) void k4_stats(const float* __restrict__ Y,
                                                const float* __restrict__ gamma,
                                                const float* __restrict__ beta,
                                                float* __restrict__ ss) {
    const int o = blockIdx.x, tid = threadIdx.x;
    float s = 0.f, s2 = 0.f;
    for (int n = 0; n < NN; n++) {
        const float* p = Y + ((size_t)n * CO + o) * COLS;
        for (int j = tid; j < COLS; j += 256) { float v = p[j]; s += v; s2 += v * v; }
    }
    __shared__ float r1[256], r2[256];
    r1[tid] = s; r2[tid] = s2; __syncthreads();
    for (int d = 128; d > 0; d >>= 1) {
        if (tid < d) { r1[tid] += r1[tid + d]; r2[tid] += r2[tid + d]; }
        __syncthreads();
    }
    if (tid == 0) {
        const float cnt = (float)NN * (float)COLS;
        float mean = r1[0] / cnt;
        float var  = r2[0] / cnt - mean * mean;
        float sc   = gamma[o] * rsqrtf(var + 1e-5f);
        ss[o] = sc; ss[CO + o] = beta[o] - mean * sc;
    }
}

// =====================================================================
// K5: fused BN apply + ReLU (in place on Y)
// =====================================================================
__global__ void k5_bnrelu(float* __restrict__ Y, const float* __restrict__ ss) {
    size_t idx = (size_t)blockIdx.x * 256 + threadIdx.x;
    const size_t total = (size_t)NN * CO * COLS;
    if (idx >= total) return;
    int o = (int)((idx / COLS) % CO);
    float v = Y[idx] * ss[o] + ss[CO + o];
    Y[idx] = fmaxf(v, 0.f);
}

// =====================================================================
extern "C" void kernel_launch(void* const* d_in, const int* in_sizes, int n_in,
                              void* d_out, int out_size, void* d_ws, size_t ws_size,
                              hipStream_t stream) {
    (void)in_sizes; (void)n_in; (void)out_size; (void)ws_size;
    const float* x     = (const float*)d_in[0];
    const float* xx    = (const float*)d_in[1];
    const float* Apow  = (const float*)d_in[2];
    const float* w_mlp = (const float*)d_in[3];
    // d_in[4] = b_mlp: unused (cancels under BatchNorm mean subtraction)
    const float* gamma = (const float*)d_in[5];
    const float* beta  = (const float*)d_in[6];
    float* Y = (float*)d_out;

    char* ws = (char*)d_ws;
    float* Bsum = (float*)ws;                               // 80,000 B
    float* AC   = (float*)(ws + 80128);                     // 1,040,000 B
    float* ss   = (float*)(ws + 80128 + 1040000);           // 3,072 B
    float* S    = (float*)(ws + 80128 + 1040000 + 3200);    // NB*832*3200*4 = 42.6 MB

    hipMemsetAsync(Bsum, 0, (size_t)NN * VV * VV * sizeof(float), stream);

    k0_adj<<<dim3(NN, TT), 256, 0, stream>>>(xx, Bsum);
    k1_ac<<<(NN * NSS * VV * VV + 255) / 256, 256, 0, stream>>>(Bsum, Apow, AC);

    for (int n0 = 0; n0 < NN; n0 += NB) {
        k2_support<<<dim3(CC, NSS, NB), 256, 0, stream>>>(x, AC, S, n0);
        k3_gemm<<<dim3(COLS / BN, CO / BM, NB), 256, 0, stream>>>(w_mlp, S, Y, n0);
    }

    k4_stats<<<CO, 256, 0, stream>>>(Y, gamma, beta, ss);
    const size_t total = (size_t)NN * CO * COLS;
    k5_bnrelu<<<(unsigned)((total + 255) / 256), 256, 0, stream>>>(Y, ss);
}